// CLNet_5557687681860
// MI455X (gfx1250) — compile-verified
//
#include <hip/hip_runtime.h>
#include <hip/hip_bf16.h>
#include <math.h>

typedef float v2f __attribute__((ext_vector_type(2)));
typedef float v8f __attribute__((ext_vector_type(8)));

__device__ __forceinline__ v8f wmma4(v2f a, v2f b, v8f c) {
  // V_WMMA_F32_16X16X4_F32 : D = A(16x4) * B(4x16) + C(16x16), all f32
  return __builtin_amdgcn_wmma_f32_16x16x4_f32(false, a, false, b, (short)0, c, false, false);
}
__device__ __forceinline__ v8f zero8() { v8f v = {0.f,0.f,0.f,0.f,0.f,0.f,0.f,0.f}; return v; }
__device__ __forceinline__ float elu1(float x) { return x > 0.f ? x : expm1f(x); }

#define UU 45
#define WHH 22
#define LL 900
#define DD 80
#define NROWS 3600   /* B*L */

// ---------------------------------------------------------------------------
// Kernel 1: per-window CNN encoder, fully fused in LDS. One block per window.
// Window fill uses CDNA5 async global->LDS copies (ASYNCcnt-tracked).
// ---------------------------------------------------------------------------
__global__ __launch_bounds__(256) void k_encoder(
    const float* __restrict__ z,
    const float* __restrict__ cw1, const float* __restrict__ cb1,
    const float* __restrict__ cw2, const float* __restrict__ cb2,
    const float* __restrict__ cw3, const float* __restrict__ cb3,
    const float* __restrict__ cw4, const float* __restrict__ cb4,
    float* __restrict__ feat)
{
  __shared__ __attribute__((aligned(16))) float win[45][80];
  __shared__ float c1[4][22][40];
  __shared__ float c2[8][11][20];
  __shared__ float c3[16][5][10];
  __shared__ float w1s[36],  b1s[4];
  __shared__ float w2s[288], b2s[8];
  __shared__ float w3s[1152], b3s[16];
  __shared__ float w4s[2304], b4s[16];

  const int tid = threadIdx.x;
  for (int i = tid; i < 36;   i += 256) w1s[i] = cw1[i];
  for (int i = tid; i < 4;    i += 256) b1s[i] = cb1[i];
  for (int i = tid; i < 288;  i += 256) w2s[i] = cw2[i];
  for (int i = tid; i < 8;    i += 256) b2s[i] = cb2[i];
  for (int i = tid; i < 1152; i += 256) w3s[i] = cw3[i];
  for (int i = tid; i < 16;   i += 256) b3s[i] = cb3[i];
  for (int i = tid; i < 2304; i += 256) w4s[i] = cw4[i];
  for (int i = tid; i < 16;   i += 256) b4s[i] = cb4[i];

  const int bid = blockIdx.x;
  const int bb = bid / LL;
  const int t  = bid % LL;
  const int end = min(t + WHH + 1, LL);
  const int lo  = max(t - WHH, 0);
  const int j0  = lo - (end - UU);        // first valid window row; rows [j0,45) real

  // zero the invalid (front) rows
  for (int i = tid; i < j0 * DD; i += 256) win[i / DD][i % DD] = 0.f;

  // async copy of the contiguous valid region: win[j0..44][*] <- z[bb, lo..end-1, *]
  {
    const int nchunks = (UU - j0) * (DD * 4 / 16);   // 16B chunks (20 per row)
    const unsigned ldsbase =
        (unsigned)(uintptr_t)(&win[0][0]) + (unsigned)(j0 * DD * 4);
    const unsigned long long gbase =
        (unsigned long long)(uintptr_t)(z + ((size_t)(bb * LL + lo)) * DD);
    for (int c = tid; c < nchunks; c += 256) {
      unsigned loff = ldsbase + (unsigned)c * 16u;
      unsigned long long ga = gbase + (unsigned long long)c * 16ull;
      asm volatile("global_load_async_to_lds_b128 %0, %1, off"
                   :: "v"(loff), "v"(ga) : "memory");
    }
    asm volatile("s_wait_asynccnt 0" ::: "memory");
  }
  __syncthreads();

  // conv1 (1->4, SAME) + ELU + maxpool 2x2 -> (4,22,40)
  for (int i = tid; i < 4 * 22 * 40; i += 256) {
    int oc = i / (22 * 40), rem = i % (22 * 40), py = rem / 40, px = rem % 40;
    float mx = -3.0e38f;
    for (int sy = 0; sy < 2; ++sy) for (int sx = 0; sx < 2; ++sx) {
      int y = py * 2 + sy, x = px * 2 + sx;
      float s = b1s[oc];
      for (int dy = -1; dy <= 1; ++dy) for (int dx = -1; dx <= 1; ++dx) {
        int yy = y + dy, xx = x + dx;
        if (yy >= 0 && yy < 45 && xx >= 0 && xx < 80)
          s += win[yy][xx] * w1s[oc * 9 + (dy + 1) * 3 + (dx + 1)];
      }
      mx = fmaxf(mx, elu1(s));
    }
    c1[oc][py][px] = mx;
  }
  __syncthreads();

  // conv2 (4->8) + ELU + pool 2x2 -> (8,11,20)
  for (int i = tid; i < 8 * 11 * 20; i += 256) {
    int oc = i / (11 * 20), rem = i % (11 * 20), py = rem / 20, px = rem % 20;
    float mx = -3.0e38f;
    for (int sy = 0; sy < 2; ++sy) for (int sx = 0; sx < 2; ++sx) {
      int y = py * 2 + sy, x = px * 2 + sx;
      float s = b2s[oc];
      for (int ic = 0; ic < 4; ++ic)
        for (int dy = -1; dy <= 1; ++dy) for (int dx = -1; dx <= 1; ++dx) {
          int yy = y + dy, xx = x + dx;
          if (yy >= 0 && yy < 22 && xx >= 0 && xx < 40)
            s += c1[ic][yy][xx] * w2s[((oc * 4 + ic) * 3 + dy + 1) * 3 + dx + 1];
        }
      mx = fmaxf(mx, elu1(s));
    }
    c2[oc][py][px] = mx;
  }
  __syncthreads();

  // conv3 (8->16) + ELU + pool 2x2 -> (16,5,10)
  for (int i = tid; i < 16 * 5 * 10; i += 256) {
    int oc = i / 50, rem = i % 50, py = rem / 10, px = rem % 10;
    float mx = -3.0e38f;
    for (int sy = 0; sy < 2; ++sy) for (int sx = 0; sx < 2; ++sx) {
      int y = py * 2 + sy, x = px * 2 + sx;
      float s = b3s[oc];
      for (int ic = 0; ic < 8; ++ic)
        for (int dy = -1; dy <= 1; ++dy) for (int dx = -1; dx <= 1; ++dx) {
          int yy = y + dy, xx = x + dx;
          if (yy >= 0 && yy < 11 && xx >= 0 && xx < 20)
            s += c2[ic][yy][xx] * w3s[((oc * 8 + ic) * 3 + dy + 1) * 3 + dx + 1];
        }
      mx = fmaxf(mx, elu1(s));
    }
    c3[oc][py][px] = mx;
  }
  __syncthreads();

  // conv4 (16->16) + ELU + pool 1x2 -> (16,5,5), write flattened feat(400)
  for (int i = tid; i < 16 * 5 * 5; i += 256) {
    int oc = i / 25, rem = i % 25, py = rem / 5, px = rem % 5;
    float mx = -3.0e38f;
    for (int sx = 0; sx < 2; ++sx) {
      int y = py, x = px * 2 + sx;
      float s = b4s[oc];
      for (int ic = 0; ic < 16; ++ic)
        for (int dy = -1; dy <= 1; ++dy) for (int dx = -1; dx <= 1; ++dx) {
          int yy = y + dy, xx = x + dx;
          if (yy >= 0 && yy < 5 && xx >= 0 && xx < 10)
            s += c3[ic][yy][xx] * w4s[((oc * 16 + ic) * 3 + dy + 1) * 3 + dx + 1];
        }
      mx = fmaxf(mx, elu1(s));
    }
    feat[bid * 400 + (oc * 5 + py) * 5 + px] = mx;
  }
}

// ---------------------------------------------------------------------------
// Kernel 2: z_t = ELU(emb([ FC2(ELU(FC1(feat))) ; onehot ]))  via f32 WMMA.
// 128 threads = 4 waves; each wave owns one M-tile of 16 rows (64 rows/block).
// Guards are branchless (clamped address + mask multiply) to keep EXEC full
// and let loads coalesce to b64.
// ---------------------------------------------------------------------------
__global__ __launch_bounds__(128) void k_mlp(
    const float* __restrict__ feat,
    const float* __restrict__ fc1w, const float* __restrict__ fc1b,
    const float* __restrict__ fc2w, const float* __restrict__ fc2b,
    const float* __restrict__ embw, const float* __restrict__ embb,
    const float* __restrict__ onehot,
    float* __restrict__ zt)
{
  __shared__ float hlds[64 * 128];
  __shared__ float zp[64 * 40];

  const int lane = threadIdx.x & 31, wid = threadIdx.x >> 5;
  const int half = lane >> 4, lr = lane & 15;
  const int rows0 = blockIdx.x * 64;
  const int mrow  = rows0 + wid * 16;

  // ---- FC1: (16 x 400) @ (400 x 128) ----
  v8f acc[8];
#pragma unroll
  for (int i = 0; i < 8; ++i) acc[i] = zero8();
  const int arow = mrow + lr;
  const float am = (arow < NROWS) ? 1.f : 0.f;
  const float* arp = feat + (size_t)((arow < NROWS) ? arow : 0) * 400;
  for (int kk = 0; kk < 100; ++kk) {
    const int k0 = kk * 4 + 2 * half;
    v2f a; a.x = arp[k0] * am; a.y = arp[k0 + 1] * am;
#pragma unroll
    for (int nt = 0; nt < 8; ++nt) {
      const int n = nt * 16 + lr;
      v2f bfr; bfr.x = fc1w[n * 400 + k0]; bfr.y = fc1w[n * 400 + k0 + 1];
      acc[nt] = wmma4(a, bfr, acc[nt]);
    }
  }
#pragma unroll
  for (int nt = 0; nt < 8; ++nt)
#pragma unroll
    for (int r = 0; r < 8; ++r) {
      const int lrow = wid * 16 + r + 8 * half;
      const int n = nt * 16 + lr;
      hlds[lrow * 128 + n] = elu1(acc[nt][r] + fc1b[n]);
    }
  __syncthreads();

  // ---- FC2: (16 x 128) @ (128 x 40), N padded to 48 ----
  v8f a2[3];
#pragma unroll
  for (int i = 0; i < 3; ++i) a2[i] = zero8();
  const float* hrp = hlds + (wid * 16 + lr) * 128;
  {
    const int n2 = 32 + lr;                    // N-tile 2 column
    const float nm[3] = { 1.f, 1.f, (n2 < 40) ? 1.f : 0.f };
    const int nc[3] = { lr, 16 + lr, (n2 < 40) ? n2 : 0 };
    for (int kk = 0; kk < 32; ++kk) {
      const int k0 = kk * 4 + 2 * half;
      v2f a; a.x = hrp[k0]; a.y = hrp[k0 + 1];
#pragma unroll
      for (int nt = 0; nt < 3; ++nt) {
        v2f bfr;
        bfr.x = fc2w[nc[nt] * 128 + k0] * nm[nt];
        bfr.y = fc2w[nc[nt] * 128 + k0 + 1] * nm[nt];
        a2[nt] = wmma4(a, bfr, a2[nt]);
      }
    }
  }
#pragma unroll
  for (int nt = 0; nt < 3; ++nt)
#pragma unroll
    for (int r = 0; r < 8; ++r) {
      const int n = nt * 16 + lr;
      if (n < 40) {
        const int lrow = wid * 16 + r + 8 * half;
        zp[lrow * 40 + n] = a2[nt][r] + fc2b[n];
      }
    }
  __syncthreads();

  // ---- embedding: z_t = ELU(embw @ [z_p ; onehot] + embb) ----
  for (int i = threadIdx.x; i < 64 * 40; i += 128) {
    const int lrow = i / 40, o = i % 40;
    const int grow = rows0 + lrow;
    if (grow < NROWS) {
      const int bb = grow / LL;
      float s = embb[o];
      for (int j = 0; j < 40; ++j) s += embw[o * 45 + j] * zp[lrow * 40 + j];
      for (int c = 0; c < 5; ++c)  s += embw[o * 45 + 40 + c] * onehot[bb * 5 + c];
      zt[grow * 40 + o] = elu1(s);
    }
  }
}

// ---------------------------------------------------------------------------
// Kernel 3a: Q/K/V = z_t @ W^T + b via f32 WMMA (K=40, N=40 padded to 48).
// ---------------------------------------------------------------------------
__global__ __launch_bounds__(128) void k_qkv(
    const float* __restrict__ zt,
    const float* __restrict__ wq, const float* __restrict__ bq,
    const float* __restrict__ wk, const float* __restrict__ bk,
    const float* __restrict__ wv, const float* __restrict__ bvv,
    float* __restrict__ qb, float* __restrict__ kb, float* __restrict__ vb)
{
  const int lane = threadIdx.x & 31, wid = threadIdx.x >> 5;
  const int half = lane >> 4, lr = lane & 15;
  const int mrow = blockIdx.x * 64 + wid * 16;
  const int arow = mrow + lr;
  const float am = (arow < NROWS) ? 1.f : 0.f;
  const float* arp = zt + (size_t)((arow < NROWS) ? arow : 0) * 40;

  v2f af[10];
#pragma unroll
  for (int kk = 0; kk < 10; ++kk) {
    const int k0 = kk * 4 + 2 * half;
    af[kk].x = arp[k0] * am;
    af[kk].y = arp[k0 + 1] * am;
  }
  const int n2 = 32 + lr;
  const float nm[3] = { 1.f, 1.f, (n2 < 40) ? 1.f : 0.f };
  const int nc[3] = { lr, 16 + lr, (n2 < 40) ? n2 : 0 };

  const float* Ws[3] = { wq, wk, wv };
  const float* Bs[3] = { bq, bk, bvv };
  float*       Os[3] = { qb, kb, vb };
#pragma unroll
  for (int s = 0; s < 3; ++s) {
    v8f acc[3];
#pragma unroll
    for (int i = 0; i < 3; ++i) acc[i] = zero8();
#pragma unroll
    for (int kk = 0; kk < 10; ++kk) {
      const int k0 = kk * 4 + 2 * half;
#pragma unroll
      for (int nt = 0; nt < 3; ++nt) {
        v2f bfr;
        bfr.x = Ws[s][nc[nt] * 40 + k0] * nm[nt];
        bfr.y = Ws[s][nc[nt] * 40 + k0 + 1] * nm[nt];
        acc[nt] = wmma4(af[kk], bfr, acc[nt]);
      }
    }
#pragma unroll
    for (int nt = 0; nt < 3; ++nt)
#pragma unroll
      for (int r = 0; r < 8; ++r) {
        const int n = nt * 16 + lr;
        const int row = mrow + r + 8 * half;
        if (n < 40 && row < NROWS) Os[s][row * 40 + n] = acc[nt][r] + Bs[s][n];
      }
  }
}

// ---------------------------------------------------------------------------
// Kernel 3b: fused masked attention + residual + 3 output heads.
// One block (8 waves) per (batch, 16-query M-tile). Scores live in LDS.
// ---------------------------------------------------------------------------
#define NKT 57          /* ceil(900/16) */
#define SCW 912         /* NKT*16 */

__global__ __launch_bounds__(256) void k_attn(
    const float* __restrict__ qb, const float* __restrict__ kb,
    const float* __restrict__ vb, const float* __restrict__ zt,
    const int* __restrict__ dur,
    const float* __restrict__ flw1, const float* __restrict__ flb1,
    const float* __restrict__ flw2, const float* __restrict__ flb2,
    const float* __restrict__ stw1, const float* __restrict__ stb1,
    const float* __restrict__ stw2, const float* __restrict__ stb2,
    const float* __restrict__ edw1, const float* __restrict__ edb1,
    const float* __restrict__ edw2, const float* __restrict__ edb2,
    float* __restrict__ out)
{
  __shared__ float sc[16 * SCW];     // exp-scores
  __shared__ float xacc[16 * 48];    // attn@V accumulators
  __shared__ float rsum[16];
  __shared__ float xrow[16 * 40];

  const int lane = threadIdx.x & 31, wid = threadIdx.x >> 5;
  const int half = lane >> 4, lr = lane & 15;
  const int bbat = blockIdx.x / NKT;
  const int mt   = blockIdx.x % NKT;
  const int m0   = mt * 16;
  const int durb = dur[bbat];
  const float ninf = -__builtin_inff();
  const float invs = 0.15811388300841898f;  // 1/sqrt(40)

  // ---- QK^T scores (f32 WMMA), scale + duration mask ----
  v2f aq[10];
  {
    const int m = m0 + lr;
    const float qm = (m < LL) ? 1.f : 0.f;
    const float* qr = qb + (size_t)(bbat * LL + ((m < LL) ? m : 0)) * 40;
#pragma unroll
    for (int kk = 0; kk < 10; ++kk) {
      const int k0 = kk * 4 + 2 * half;
      aq[kk].x = qr[k0] * qm;
      aq[kk].y = qr[k0 + 1] * qm;
    }
  }
  for (int nt = wid; nt < NKT; nt += 8) {
    const int kbase = nt * 16;
    const int key = kbase + lr;
    const float km = (key < LL) ? 1.f : 0.f;
    const float* kr = kb + (size_t)(bbat * LL + ((key < LL) ? key : 0)) * 40;
    v8f s = zero8();
#pragma unroll
    for (int kk = 0; kk < 10; ++kk) {
      const int k0 = kk * 4 + 2 * half;
      v2f bfr; bfr.x = kr[k0] * km; bfr.y = kr[k0 + 1] * km;
      s = wmma4(aq[kk], bfr, s);
    }
#pragma unroll
    for (int r = 0; r < 8; ++r) {
      const int lm = r + 8 * half;
      const int m = m0 + lm;
      float val;
      if (m < LL && key < LL) {
        val = s[r] * invs;
        if (!((m < durb) && (key < durb))) val = -1e-12f;   // reference mask value
      } else val = ninf;                                     // padding columns/rows
      sc[lm * SCW + kbase + lr] = val;
    }
  }
  __syncthreads();

  // ---- row softmax (unnormalized exp; keep the row sum for the epilogue) ----
  for (int lm = wid; lm < 16; lm += 8) {
    float* row = sc + lm * SCW;
    if (m0 + lm < LL) {
      float mx = ninf;
      for (int c = lane; c < SCW; c += 32) mx = fmaxf(mx, row[c]);
#pragma unroll
      for (int o = 16; o >= 1; o >>= 1) mx = fmaxf(mx, __shfl_xor(mx, o, 32));
      float sm = 0.f;
      for (int c = lane; c < SCW; c += 32) { float e = __expf(row[c] - mx); row[c] = e; sm += e; }
#pragma unroll
      for (int o = 16; o >= 1; o >>= 1) sm += __shfl_xor(sm, o, 32);
      if (lane == 0) rsum[lm] = sm;
    } else {
      for (int c = lane; c < SCW; c += 32) row[c] = 0.f;
      if (lane == 0) rsum[lm] = 1.f;
    }
  }
  for (int i = threadIdx.x; i < 16 * 48; i += 256) xacc[i] = 0.f;
  __syncthreads();

  // ---- attn @ V (f32 WMMA, K split across waves, LDS float-atomic reduce) ----
  v8f acc[3];
#pragma unroll
  for (int i = 0; i < 3; ++i) acc[i] = zero8();
  const int n2 = 32 + lr;
  const float nmv = (n2 < 40) ? 1.f : 0.f;
  const int ncv[3] = { lr, 16 + lr, (n2 < 40) ? n2 : 0 };
  const float nmm[3] = { 1.f, 1.f, nmv };
  for (int kt = wid; kt < NKT; kt += 8) {
#pragma unroll
    for (int q4 = 0; q4 < 4; ++q4) {
      const int k0 = kt * 16 + q4 * 4 + 2 * half;
      v2f a; a.x = sc[lr * SCW + k0]; a.y = sc[lr * SCW + k0 + 1];
      const float km0 = (k0 < LL) ? 1.f : 0.f;
      const float km1 = ((k0 + 1) < LL) ? 1.f : 0.f;
      const float* v0 = vb + (size_t)(bbat * LL + ((k0 < LL) ? k0 : 0)) * 40;
      const float* v1 = vb + (size_t)(bbat * LL + (((k0 + 1) < LL) ? k0 + 1 : 0)) * 40;
#pragma unroll
      for (int nt = 0; nt < 3; ++nt) {
        v2f bfr;
        bfr.x = v0[ncv[nt]] * (nmm[nt] * km0);
        bfr.y = v1[ncv[nt]] * (nmm[nt] * km1);
        acc[nt] = wmma4(a, bfr, acc[nt]);
      }
    }
  }
#pragma unroll
  for (int nt = 0; nt < 3; ++nt)
#pragma unroll
    for (int r = 0; r < 8; ++r) {
      const int lm = r + 8 * half;
      atomicAdd(&xacc[lm * 48 + nt * 16 + lr], acc[nt][r]);
    }
  __syncthreads();

  // ---- x = attn@V / rowsum + z_t ----
  for (int i = threadIdx.x; i < 16 * 40; i += 256) {
    const int lm = i / 40, d = i % 40;
    const int m = m0 + lm;
    float v = 0.f;
    if (m < LL) v = xacc[lm * 48 + d] / rsum[lm] + zt[(size_t)(bbat * LL + m) * 40 + d];
    xrow[lm * 40 + d] = v;
  }
  __syncthreads();

  // ---- three heads: 40 -> ELU(10) -> 1 ----
  if (threadIdx.x < 48) {
    const int lm = threadIdx.x / 3, hd = threadIdx.x % 3;
    const int m = m0 + lm;
    if (m < LL) {
      const float* w1 = hd == 0 ? flw1 : (hd == 1 ? stw1 : edw1);
      const float* b1 = hd == 0 ? flb1 : (hd == 1 ? stb1 : edb1);
      const float* w2 = hd == 0 ? flw2 : (hd == 1 ? stw2 : edw2);
      const float* b2 = hd == 0 ? flb2 : (hd == 1 ? stb2 : edb2);
      float o = b2[0];
      for (int i = 0; i < 10; ++i) {
        float h = b1[i];
        for (int j = 0; j < 40; ++j) h += w1[i * 40 + j] * xrow[lm * 40 + j];
        o += w2[i] * elu1(h);
      }
      out[hd * NROWS + bbat * LL + m] = o;
    }
  }
}

// ---------------------------------------------------------------------------
extern "C" void kernel_launch(void* const* d_in, const int* in_sizes, int n_in,
                              void* d_out, int out_size, void* d_ws, size_t ws_size,
                              hipStream_t stream) {
  const float* z      = (const float*)d_in[0];
  const float* onehot = (const float*)d_in[1];
  const int*   dur    = (const int*)  d_in[2];
  const float* cw1 = (const float*)d_in[3];   const float* cb1 = (const float*)d_in[4];
  const float* cw2 = (const float*)d_in[5];   const float* cb2 = (const float*)d_in[6];
  const float* cw3 = (const float*)d_in[7];   const float* cb3 = (const float*)d_in[8];
  const float* cw4 = (const float*)d_in[9];   const float* cb4 = (const float*)d_in[10];
  const float* fc1w = (const float*)d_in[11]; const float* fc1b = (const float*)d_in[12];
  const float* fc2w = (const float*)d_in[13]; const float* fc2b = (const float*)d_in[14];
  const float* embw = (const float*)d_in[15]; const float* embb = (const float*)d_in[16];
  const float* wq = (const float*)d_in[17];   const float* bq = (const float*)d_in[18];
  const float* wk = (const float*)d_in[19];   const float* bk = (const float*)d_in[20];
  const float* wv = (const float*)d_in[21];   const float* bv = (const float*)d_in[22];
  const float* flw1 = (const float*)d_in[23]; const float* flb1 = (const float*)d_in[24];
  const float* flw2 = (const float*)d_in[25]; const float* flb2 = (const float*)d_in[26];
  const float* stw1 = (const float*)d_in[27]; const float* stb1 = (const float*)d_in[28];
  const float* stw2 = (const float*)d_in[29]; const float* stb2 = (const float*)d_in[30];
  const float* edw1 = (const float*)d_in[31]; const float* edb1 = (const float*)d_in[32];
  const float* edw2 = (const float*)d_in[33]; const float* edb2 = (const float*)d_in[34];
  float* out = (float*)d_out;

  // workspace layout (floats): feat | zt | q | k | v  == ~8.06 MB
  float* feat = (float*)d_ws;
  float* zt = feat + NROWS * 400;
  float* qb = zt + NROWS * 40;
  float* kb = qb + NROWS * 40;
  float* vb = kb + NROWS * 40;

  k_encoder<<<NROWS, 256, 0, stream>>>(z, cw1, cb1, cw2, cb2, cw3, cb3, cw4, cb4, feat);
  k_mlp<<<(NROWS + 63) / 64, 128, 0, stream>>>(feat, fc1w, fc1b, fc2w, fc2b, embw, embb, onehot, zt);
  k_qkv<<<(NROWS + 63) / 64, 128, 0, stream>>>(zt, wq, bq, wk, bk, wv, bv, qb, kb, vb);
  k_attn<<<4 * NKT, 256, 0, stream>>>(qb, kb, vb, zt, dur,
                                      flw1, flb1, flw2, flb2,
                                      stw1, stb1, stw2, stb2,
                                      edw1, edb1, edw2, edb2, out);
}